// DDK_77644418777662
// MI455X (gfx1250) — compile-verified
//
#include <hip/hip_runtime.h>
#include <stdint.h>

// Problem constants (fixed by the reference): rows = l*p*m = 32768, n = 1024.
#define N_K        1024                // recurrence length per row
#define TK         32                  // k-values per tile (128 B per row)
#define NTILES     (N_K / TK)          // 32 tiles
#define RPB        128                 // rows per block == threads per block (4 wave32)
#define LDS_STRIDE 36                  // floats per row in LDS (32 + 4 pad; conflict-free b128 reads)
#define CHUNKS     8                   // 16B async chunks per thread per tile

typedef __attribute__((ext_vector_type(4))) float v4f;

// w_next = w + a/w - b*x  ==  fma(-b, x, fma(a, rcp(w), w))
__device__ __forceinline__ float wstep(float w, float x, float a, float nb) {
  float r = __builtin_amdgcn_rcpf(w);          // v_rcp_f32 (avoid IEEE div sequence)
  return __builtin_fmaf(nb, x, __builtin_fmaf(a, r, w));
}

// Coalesced async copy of one 128-row x 32-float tile into LDS.
// chunk = j*RPB + tid ; row = chunk>>3 ; sub = chunk&7  => lanes 8t..8t+7 cover
// one contiguous 128B row segment (fully coalesced per 8 lanes).
__device__ __forceinline__ void async_tile_load(const float* __restrict__ in,
                                                uint32_t lds_byte_base,
                                                int row0, int ktile, int tid) {
#pragma unroll
  for (int j = 0; j < CHUNKS; ++j) {
    const int chunk = j * RPB + tid;
    const int row   = chunk >> 3;
    const int sub   = chunk & 7;
    const char* gsrc =
        (const char*)(in + (size_t)(row0 + row) * N_K + (size_t)ktile * TK) + sub * 16;
    const uint32_t ldst =
        lds_byte_base + (uint32_t)(row * (LDS_STRIDE * 4) + sub * 16);
    asm volatile("global_load_async_to_lds_b128 %0, %1, off"
                 :: "v"(ldst), "v"(gsrc)
                 : "memory");
  }
}

template <int N>
__device__ __forceinline__ void wait_asynccnt() {
  asm volatile("s_wait_asynccnt %0" :: "i"(N) : "memory");
}

__global__ __launch_bounds__(RPB) void DDK_77644418777662_scan(
    const float* __restrict__ in, const float* __restrict__ alpha,
    const float* __restrict__ beta, float* __restrict__ out) {
  __shared__ float lds[2][RPB * LDS_STRIDE];   // 36 KB double buffer

  const int tid  = threadIdx.x;
  const int row0 = blockIdx.x * RPB;
  const float a  = alpha[0];
  const float nb = -beta[0];

  const uint32_t l0 = (uint32_t)(uintptr_t)&lds[0][0];  // LDS byte offsets
  const uint32_t l1 = (uint32_t)(uintptr_t)&lds[1][0];

  // Prefetch tile 0.
  async_tile_load(in, l0, row0, 0, tid);

  float* __restrict__ orow = out + (size_t)(row0 + tid) * N_K;
  float w = 1.0f;       // w[0]
  float xprev = 0.0f;   // carry x[4G-1] across quads (unused for G==0)

  for (int c = 0; c < NTILES; ++c) {
    const float* cur = (c & 1) ? &lds[1][0] : &lds[0][0];
    if (c + 1 < NTILES) {
      // Issue next tile into the other buffer (safe: previous iteration's
      // trailing barrier guarantees everyone is done reading it).
      async_tile_load(in, (c & 1) ? l0 : l1, row0, c + 1, tid);
      wait_asynccnt<CHUNKS>();   // this wave's current-tile loads complete
    } else {
      wait_asynccnt<0>();
    }
    __syncthreads();             // all waves' current-tile loads complete

    const v4f* lrow = (const v4f*)(cur + tid * LDS_STRIDE);
#pragma unroll
    for (int q = 0; q < TK / 4; ++q) {
      const v4f x = lrow[q];                 // ds_load_b128, conflict-free
      const int G = c * (TK / 4) + q;        // output quad index
      float o0;
      if (G == 0) {
        o0 = 1.0f;                           // out[0] = w[0] = 1
      } else {
        w = wstep(w, xprev, a, nb);          // out[4G] uses x[4G-1] (carry)
        o0 = w;
      }
      w = wstep(w, x.x, a, nb);  const float o1 = w;   // out[4G+1]
      w = wstep(w, x.y, a, nb);  const float o2 = w;   // out[4G+2]
      w = wstep(w, x.z, a, nb);  const float o3 = w;   // out[4G+3]
      xprev = x.w;                                     // carry x[4G+3]
      v4f o = {o0, o1, o2, o3};
      __builtin_nontemporal_store(o, (v4f*)orow + G);  // aligned 16B NT store
    }
    __syncthreads();             // protect cur buffer before next async overwrite
  }
}

extern "C" void kernel_launch(void* const* d_in, const int* in_sizes, int n_in,
                              void* d_out, int out_size, void* d_ws, size_t ws_size,
                              hipStream_t stream) {
  const float* in    = (const float*)d_in[0];
  const float* alpha = (const float*)d_in[1];
  const float* beta  = (const float*)d_in[2];
  float* out = (float*)d_out;

  const int rows = in_sizes[0] / N_K;   // 8*1*4096 = 32768
  dim3 grid(rows / RPB), block(RPB);    // 256 blocks x 128 threads
  DDK_77644418777662_scan<<<grid, block, 0, stream>>>(in, alpha, beta, out);
}